// SolveSchedulingQP_62826781606433
// MI455X (gfx1250) — compile-verified
//
#include <hip/hip_runtime.h>
#include <stdint.h>

constexpr int kN = 24;          // horizon
constexpr int kM = 46;          // 2*(N-1) constraints
constexpr int kIters = 25;
constexpr float kRamp = 0.4f;
constexpr float kSigma = 0.1f;
constexpr int TPB = 64;         // 2 waves per block

__device__ __forceinline__ float frcp(float x) { return __builtin_amdgcn_rcpf(x); }

__global__ __launch_bounds__(TPB) void qp_ipm_kernel(
    const float* __restrict__ g_z0,
    const float* __restrict__ g_mu,
    const float* __restrict__ g_dg,
    const float* __restrict__ g_d2g,
    float* __restrict__ g_out,
    int nprob)
{
  // Staging buffer: 4 arrays x TPB rows x 24 floats = 24 KB LDS
  __shared__ float smem[4 * TPB * kN];

  const int tid = threadIdx.x;
  const int blockBase = blockIdx.x * TPB;
  const int validProbs = min(TPB, nprob - blockBase);
  // last valid 16B chunk start within this block's slab (per array)
  const uint32_t lim = (uint32_t)(validProbs * kN * 4) - 16u;

  const float* bases[4] = {
      g_z0 + (size_t)blockBase * kN,
      g_mu + (size_t)blockBase * kN,
      g_dg + (size_t)blockBase * kN,
      g_d2g + (size_t)blockBase * kN };

  // ---- CDNA5 async copy engine: coalesced global -> LDS, 16B per lane ----
  // OOB lanes are clamped to the last valid chunk (benign duplicate write
  // of the same chunk to the same LDS address) -> no EXEC branching.
  #pragma unroll
  for (int a = 0; a < 4; ++a) {
    uint32_t ldsArr = (uint32_t)(uintptr_t)(&smem[a * TPB * kN]);
    #pragma unroll
    for (int k = 0; k < (kN * 4) / 16; ++k) {   // 6 chunks of 16B per lane
      uint32_t goff = (uint32_t)((k * TPB + tid) * 16);
      uint32_t eff  = goff <= lim ? goff : lim;
      uint32_t laddr = ldsArr + eff;
      asm volatile("global_load_async_to_lds_b128 %0, %1, %2"
                   :: "v"(laddr), "v"(eff), "s"(bases[a])
                   : "memory");
    }
  }
  asm volatile("s_wait_asynccnt 0" ::: "memory");
  __syncthreads();

  if (blockBase + tid >= nprob) return;

  const float* rz0 = &smem[0 * TPB * kN + tid * kN];
  const float* rmu = &smem[1 * TPB * kN + tid * kN];
  const float* rdg = &smem[2 * TPB * kN + tid * kN];
  const float* rd2 = &smem[3 * TPB * kN + tid * kN];

  float q[kN], p[kN], z[kN];
  #pragma unroll
  for (int i = 0; i < kN; ++i) {
    float d2 = rd2[i];
    q[i] = d2 + 1.0f;
    p[i] = rdg[i] - d2 * rz0[i] - rmu[i];
    z[i] = -p[i] * frcp(q[i]);              // unconstrained minimizer
  }
  float s[kM], lam[kM];
  #pragma unroll
  for (int j = 0; j < kM; ++j) { s[j] = 1.0f; lam[j] = 1.0f; }

  #pragma unroll 1
  for (int it = 0; it < kIters; ++it) {
    // centering parameter: sigma * mean(s .* lam)
    float acc = 0.0f;
    #pragma unroll
    for (int j = 0; j < kM; ++j) acc += s[j] * lam[j];
    const float sigmu = (kSigma / (float)kM) * acc;

    float w[kM], rp[kM], rcs[kM];           // w = lam/s, r_prim, r_cent/s
    #pragma unroll
    for (int j = 0; j < kM; ++j) {
      float is = frcp(s[j]);
      w[j] = lam[j] * is;
      float gz;
      if (j < kN - 1) gz = z[j] - z[j + 1];                    // +D rows
      else            gz = z[j - kN + 2] - z[j - kN + 1];      // -D rows
      rp[j]  = gz + s[j] - kRamp;
      rcs[j] = lam[j] - sigmu * is;         // (s*lam - sigmu)/s
    }

    // Combined difference coefficients for rhs:
    // (lam @ G)_i + ((w*rp - rcs) @ G)_i  =  dc[i] - dc[i-1]
    float dc[kN - 1];
    #pragma unroll
    for (int k = 0; k < kN - 1; ++k) {
      float t1 = w[k] * rp[k] - rcs[k];
      float t2 = w[k + kN - 1] * rp[k + kN - 1] - rcs[k + kN - 1];
      dc[k] = (lam[k] - lam[k + kN - 1]) + (t1 - t2);
    }
    float rhs[kN];
    #pragma unroll
    for (int i = 0; i < kN; ++i) {
      float d = 0.0f;
      if (i < kN - 1) d += dc[i];
      if (i > 0)      d -= dc[i - 1];
      rhs[i] = -(q[i] * z[i] + p[i] + d);
    }

    // K = diag(q) + G^T W G is SPD TRIDIAGONAL:
    //   edge weight v[k] = w[k] + w[k+23]; diag = q_i + v_{i-1} + v_i; off = -v_i
    float v[kN - 1];
    #pragma unroll
    for (int k = 0; k < kN - 1; ++k) v[k] = w[k] + w[k + kN - 1];

    // Thomas forward sweep
    float cp[kN - 1], xp[kN];
    {
      float m0 = q[0] + v[0];
      float im = frcp(m0);
      cp[0] = -v[0] * im;
      xp[0] = rhs[0] * im;
      #pragma unroll
      for (int i = 1; i < kN; ++i) {
        float e  = -v[i - 1];
        float di = q[i] + v[i - 1] + ((i < kN - 1) ? v[i] : 0.0f);
        float mm = di - e * cp[i - 1];
        float im2 = frcp(mm);
        if (i < kN - 1) cp[i] = -v[i] * im2;
        xp[i] = (rhs[i] - e * xp[i - 1]) * im2;
      }
    }
    // back substitution
    float dz[kN];
    dz[kN - 1] = xp[kN - 1];
    #pragma unroll
    for (int i = kN - 2; i >= 0; --i) dz[i] = xp[i] - cp[i] * dz[i + 1];

    // dlam, ds, and fused fraction-to-boundary step size
    float dlam[kM], dsv[kM];
    float mr0 = 1e9f, mr1 = 1e9f, mr2 = 1e9f, mr3 = 1e9f;
    #pragma unroll
    for (int j = 0; j < kM; ++j) {
      float gdz;
      if (j < kN - 1) gdz = dz[j] - dz[j + 1];
      else            gdz = dz[j - kN + 2] - dz[j - kN + 1];
      float dl  = w[j] * (gdz + rp[j]) - rcs[j];
      float dsj = -s[j] * (rcs[j] + dl) * frcp(lam[j]);
      dlam[j] = dl;
      dsv[j]  = dsj;
      float r1 = s[j] * frcp(-dsj);
      r1 = (dsj < 0.0f) ? r1 : 1e9f;
      float r2 = lam[j] * frcp(-dl);
      r2 = (dl < 0.0f) ? r2 : 1e9f;
      float r = fminf(r1, r2);
      if ((j & 3) == 0)      mr0 = fminf(mr0, r);
      else if ((j & 3) == 1) mr1 = fminf(mr1, r);
      else if ((j & 3) == 2) mr2 = fminf(mr2, r);
      else                   mr3 = fminf(mr3, r);
    }
    float minrat = fminf(fminf(mr0, mr1), fminf(mr2, mr3));
    float alpha = fminf(1.0f, 0.99f * minrat);

    #pragma unroll
    for (int i = 0; i < kN; ++i) z[i] += alpha * dz[i];
    #pragma unroll
    for (int j = 0; j < kM; ++j) {
      s[j]   += alpha * dsv[j];
      lam[j] += alpha * dlam[j];
    }
  }

  // write result row (96B per thread, 16B-aligned -> b128 stores)
  float4* o = reinterpret_cast<float4*>(g_out + (size_t)(blockBase + tid) * kN);
  #pragma unroll
  for (int i = 0; i < kN / 4; ++i)
    o[i] = make_float4(z[4 * i + 0], z[4 * i + 1], z[4 * i + 2], z[4 * i + 3]);
}

extern "C" void kernel_launch(void* const* d_in, const int* in_sizes, int n_in,
                              void* d_out, int out_size, void* d_ws, size_t ws_size,
                              hipStream_t stream) {
  const float* z0  = (const float*)d_in[0];
  const float* mu  = (const float*)d_in[1];
  const float* dg  = (const float*)d_in[2];
  const float* d2g = (const float*)d_in[3];
  float* out = (float*)d_out;

  const int nprob = in_sizes[0] / kN;   // B = 8192
  const int grid = (nprob + TPB - 1) / TPB;
  qp_ipm_kernel<<<grid, TPB, 0, stream>>>(z0, mu, dg, d2g, out, nprob);
}